// LstmEncDeltaStackedFullPred_33698313405135
// MI455X (gfx1250) — compile-verified
//
#include <hip/hip_runtime.h>
#include <hip/hip_bf16.h>
#include <math.h>

typedef float v2f __attribute__((ext_vector_type(2)));
typedef float v8f __attribute__((ext_vector_type(8)));

#define NB 1024         // batch
#define INP_STRIDE 192  // 180 catted + 2 'a' + 10 pad(zero)
#define GWP_STRIDE 184  // gw padded K (182 -> 184, zeros)

__device__ __forceinline__ float sigm(float x) { return 1.0f / (1.0f + expf(-x)); }

// ---------------------------------------------------------------------------
// Generic LSTM cell (hidden=10), weights in LDS. Gate order: i, f, g, o.
// ---------------------------------------------------------------------------
__device__ __forceinline__ void lstm_cell(const float* W, int xin, const float* U,
                                          const float* bi, const float* bh,
                                          const float* x, float* h, float* c) {
  float g[40];
  for (int gg = 0; gg < 40; ++gg) {
    float acc = bi[gg] + bh[gg];
    for (int k = 0; k < xin; ++k) acc += x[k] * W[gg * xin + k];
#pragma unroll
    for (int k = 0; k < 10; ++k) acc += h[k] * U[gg * 10 + k];
    g[gg] = acc;
  }
#pragma unroll
  for (int k = 0; k < 10; ++k) {
    float ig = sigm(g[k]), fg = sigm(g[10 + k]);
    float cg = tanhf(g[20 + k]), og = sigm(g[30 + k]);
    c[k] = fg * c[k] + ig * cg;
    h[k] = og * tanhf(c[k]);
  }
}

// ---------------------------------------------------------------------------
// K0: pad gw (384x182) -> gwp (384x184, cols 182..183 = 0)
// ---------------------------------------------------------------------------
__global__ __launch_bounds__(256) void pad_gw_kernel(const float* __restrict__ gw,
                                                     float* __restrict__ gwp) {
  int idx = blockIdx.x * 256 + threadIdx.x;  // 0 .. 384*184-1
  if (idx < 384 * GWP_STRIDE) {
    int n = idx / GWP_STRIDE, k = idx % GWP_STRIDE;
    gwp[idx] = (k < 182) ? gw[n * 182 + k] : 0.0f;
  }
}

// ---------------------------------------------------------------------------
// K1: scene BiLSTM (in=4) -> catted cols [t*20 .. t*20+19], t=0..7
//     + a0 = cur_state @ aw.T + ab -> inp cols 180,181 ; zero cols 182..191
// ---------------------------------------------------------------------------
__global__ __launch_bounds__(256) void scene_kernel(
    const float* __restrict__ scene, const float* __restrict__ nw_f,
    const float* __restrict__ nu_f, const float* __restrict__ nbi_f,
    const float* __restrict__ nbh_f, const float* __restrict__ nw_b,
    const float* __restrict__ nu_b, const float* __restrict__ nbi_b,
    const float* __restrict__ nbh_b, const float* __restrict__ aw,
    const float* __restrict__ ab, float* __restrict__ inp) {
  __shared__ float wf[160], uf[400], bif[40], bhf[40];
  __shared__ float wb[160], ub[400], bib[40], bhb[40];
  int tid = threadIdx.x;
  for (int i = tid; i < 160; i += 256) { wf[i] = nw_f[i]; wb[i] = nw_b[i]; }
  for (int i = tid; i < 400; i += 256) { uf[i] = nu_f[i]; ub[i] = nu_b[i]; }
  for (int i = tid; i < 40; i += 256) {
    bif[i] = nbi_f[i]; bhf[i] = nbh_f[i]; bib[i] = nbi_b[i]; bhb[i] = nbh_b[i];
  }
  __syncthreads();

  int b = blockIdx.x * 256 + tid;
  float* row = inp + (size_t)b * INP_STRIDE;

  float h[10], c[10], x[4];
  // forward
#pragma unroll
  for (int k = 0; k < 10; ++k) { h[k] = 0.f; c[k] = 0.f; }
  for (int t = 0; t < 8; ++t) {
#pragma unroll
    for (int k = 0; k < 4; ++k) x[k] = scene[b * 32 + t * 4 + k];
    lstm_cell(wf, 4, uf, bif, bhf, x, h, c);
#pragma unroll
    for (int k = 0; k < 10; ++k) row[t * 20 + k] = h[k];
  }
  // backward
#pragma unroll
  for (int k = 0; k < 10; ++k) { h[k] = 0.f; c[k] = 0.f; }
  for (int t = 7; t >= 0; --t) {
#pragma unroll
    for (int k = 0; k < 4; ++k) x[k] = scene[b * 32 + t * 4 + k];
    lstm_cell(wb, 4, ub, bib, bhb, x, h, c);
#pragma unroll
    for (int k = 0; k < 10; ++k) row[t * 20 + 10 + k] = h[k];
  }
  // a0
  float cs[4];
#pragma unroll
  for (int k = 0; k < 4; ++k) cs[k] = scene[b * 32 + 28 + k];
#pragma unroll
  for (int j = 0; j < 2; ++j) {
    float a = ab[j];
#pragma unroll
    for (int k = 0; k < 4; ++k) a += cs[k] * aw[j * 4 + k];
    row[180 + j] = a;
  }
#pragma unroll
  for (int k = 182; k < INP_STRIDE; ++k) row[k] = 0.f;
}

// ---------------------------------------------------------------------------
// K2: edge BiLSTM over deltas (in=2, 1024 timesteps fwd; 1 step bwd)
//     -> catted cols 160..179
// ---------------------------------------------------------------------------
__global__ __launch_bounds__(256) void edge_kernel(
    const float* __restrict__ scene, const float* __restrict__ ew_f,
    const float* __restrict__ eu_f, const float* __restrict__ ebi_f,
    const float* __restrict__ ebh_f, const float* __restrict__ ew_b,
    const float* __restrict__ eu_b, const float* __restrict__ ebi_b,
    const float* __restrict__ ebh_b, float* __restrict__ inp) {
  __shared__ float px[NB], py[NB];
  __shared__ float wf[80], uf[400], bif[40], bhf[40];
  __shared__ float wb[80], ub[400], bib[40], bhb[40];
  int tid = threadIdx.x;
  for (int i = tid; i < 80; i += 256) { wf[i] = ew_f[i]; wb[i] = ew_b[i]; }
  for (int i = tid; i < 400; i += 256) { uf[i] = eu_f[i]; ub[i] = eu_b[i]; }
  for (int i = tid; i < 40; i += 256) {
    bif[i] = ebi_f[i]; bhf[i] = ebh_f[i]; bib[i] = ebi_b[i]; bhb[i] = ebh_b[i];
  }
  for (int i = tid; i < NB; i += 256) {
    px[i] = scene[i * 32 + 28];
    py[i] = scene[i * 32 + 29];
  }
  __syncthreads();

  int b = blockIdx.x * 256 + tid;
  float pbx = px[b], pby = py[b];
  float h[10], c[10], g[40];
#pragma unroll
  for (int k = 0; k < 10; ++k) { h[k] = 0.f; c[k] = 0.f; }

  for (int t = 0; t < NB; ++t) {
    float x0 = px[t] - pbx, x1 = py[t] - pby;
    for (int gg = 0; gg < 40; ++gg) {
      float acc = bif[gg] + bhf[gg] + x0 * wf[gg * 2] + x1 * wf[gg * 2 + 1];
#pragma unroll
      for (int k = 0; k < 10; ++k) acc += h[k] * uf[gg * 10 + k];
      g[gg] = acc;
    }
#pragma unroll
    for (int k = 0; k < 10; ++k) {
      float ig = sigm(g[k]), fg = sigm(g[10 + k]);
      float cg = tanhf(g[20 + k]), og = sigm(g[30 + k]);
      c[k] = fg * c[k] + ig * cg;
      h[k] = og * tanhf(c[k]);
    }
  }
  float* row = inp + (size_t)b * INP_STRIDE;
#pragma unroll
  for (int k = 0; k < 10; ++k) row[160 + k] = h[k];

  // backward direction: output at original index T-1 == 1 step from zero state
  {
    float x0 = px[NB - 1] - pbx, x1 = py[NB - 1] - pby;
#pragma unroll
    for (int k = 0; k < 10; ++k) {
      float gi = bib[k] + bhb[k] + x0 * wb[k * 2] + x1 * wb[k * 2 + 1];
      float gg = bib[20 + k] + bhb[20 + k] + x0 * wb[(20 + k) * 2] + x1 * wb[(20 + k) * 2 + 1];
      float go = bib[30 + k] + bhb[30 + k] + x0 * wb[(30 + k) * 2] + x1 * wb[(30 + k) * 2 + 1];
      float c2 = sigm(gi) * tanhf(gg);  // c_prev = 0
      row[170 + k] = sigm(go) * tanhf(c2);
    }
  }
}

// ---------------------------------------------------------------------------
// K3: state0 = catted @ sw.T + sb   (1024x180)*(180x128) via WMMA f32 16x16x4
// K=180 = 45*4 exactly -> guard-free inner loop.
// ---------------------------------------------------------------------------
__global__ __launch_bounds__(256) void state0_kernel(
    const float* __restrict__ inp, const float* __restrict__ sw,
    const float* __restrict__ sb, float* __restrict__ state) {
  const int lane = threadIdx.x & 31;
  const int wave = threadIdx.x >> 5;
  const int half = lane >> 4, l16 = lane & 15;
  const int r0 = blockIdx.x * 16;
  const int m = r0 + l16;         // A row for this lane
  const int n = wave * 16 + l16;  // B/C col for this lane

  const float* ap = inp + (size_t)m * INP_STRIDE + half * 2;
  const float* bp = sw + (size_t)n * 180 + half * 2;
  v8f acc = {};
#pragma unroll 5
  for (int kt = 0; kt < 180; kt += 4) {
    v2f a, bb;
    a.x = ap[kt];  a.y = ap[kt + 1];
    bb.x = bp[kt]; bb.y = bp[kt + 1];
    acc = __builtin_amdgcn_wmma_f32_16x16x4_f32(false, a, false, bb, (short)0,
                                                acc, false, false);
  }
  float bias = sb[n];
#pragma unroll
  for (int r = 0; r < 8; ++r) {
    int rowi = r0 + r + half * 8;
    state[(size_t)rowi * 128 + n] = acc[r] + bias;
  }
}

// ---------------------------------------------------------------------------
// K4 (x12): fused GRU step.
// Block = 16 rows, 384 threads = 12 waves.
//   gi = inp@gwp.T + gbi (16x384, K=184 padded, guard-free)
//   gh = h@gu.T + gbh    (16x384, K=128, guard-free)
//   -> LDS; GRU elementwise -> h_new; heads (6x128) -> outputs + a_{s+1}.
// ---------------------------------------------------------------------------
__global__ __launch_bounds__(384) void gru_step_kernel(
    float* __restrict__ inp, const float* __restrict__ gwp,
    const float* __restrict__ gu, const float* __restrict__ gbi,
    const float* __restrict__ gbh, const float* __restrict__ hin,
    float* __restrict__ hout, const float* __restrict__ pw,
    const float* __restrict__ pb, const float* __restrict__ mw,
    const float* __restrict__ mb, const float* __restrict__ lw,
    const float* __restrict__ lb, const float* __restrict__ cw,
    const float* __restrict__ cb, const float* __restrict__ eps,
    float* __restrict__ out, int step) {
  __shared__ float s_gi[16 * 384];
  __shared__ float s_gh[16 * 384];
  __shared__ float s_h[16 * 128];
  __shared__ float s_head[16 * 6];

  const int tid = threadIdx.x;
  const int lane = tid & 31;
  const int wave = tid >> 5;  // 0..11
  const int half = lane >> 4, l16 = lane & 15;
  const int r0 = blockIdx.x * 16;
  const int m = r0 + l16;

  const float* ap_i = inp + (size_t)m * INP_STRIDE + half * 2;  // cols 182..191 zero
  const float* ap_h = hin + (size_t)m * 128 + half * 2;

  // ---- gi: 24 col-tiles, 2 per wave (K = 184 padded, guard-free) ----
#pragma unroll
  for (int tt = 0; tt < 2; ++tt) {
    const int n = (wave + tt * 12) * 16 + l16;
    const float* bp = gwp + (size_t)n * GWP_STRIDE + half * 2;
    v8f acc = {};
#pragma unroll 2
    for (int kt = 0; kt < 184; kt += 4) {
      v2f a, bb;
      a.x = ap_i[kt];  a.y = ap_i[kt + 1];
      bb.x = bp[kt];   bb.y = bp[kt + 1];
      acc = __builtin_amdgcn_wmma_f32_16x16x4_f32(false, a, false, bb, (short)0,
                                                  acc, false, false);
    }
    float bias = gbi[n];
#pragma unroll
    for (int r = 0; r < 8; ++r) s_gi[(r + half * 8) * 384 + n] = acc[r] + bias;
  }

  // ---- gh: 24 col-tiles, 2 per wave (K = 128, guard-free) ----
#pragma unroll
  for (int tt = 0; tt < 2; ++tt) {
    const int n = (wave + tt * 12) * 16 + l16;
    const float* bp = gu + (size_t)n * 128 + half * 2;
    v8f acc = {};
#pragma unroll 4
    for (int kt = 0; kt < 128; kt += 4) {
      v2f a, bb;
      a.x = ap_h[kt];  a.y = ap_h[kt + 1];
      bb.x = bp[kt];   bb.y = bp[kt + 1];
      acc = __builtin_amdgcn_wmma_f32_16x16x4_f32(false, a, false, bb, (short)0,
                                                  acc, false, false);
    }
    float bias = gbh[n];
#pragma unroll
    for (int r = 0; r < 8; ++r) s_gh[(r + half * 8) * 384 + n] = acc[r] + bias;
  }
  __syncthreads();

  // ---- GRU elementwise: 16x128 values ----
  for (int e = tid; e < 16 * 128; e += 384) {
    int mr = e >> 7, j = e & 127;
    int rowi = r0 + mr;
    float gir = s_gi[mr * 384 + j],       ghr = s_gh[mr * 384 + j];
    float giz = s_gi[mr * 384 + 128 + j], ghz = s_gh[mr * 384 + 128 + j];
    float gin = s_gi[mr * 384 + 256 + j], ghn = s_gh[mr * 384 + 256 + j];
    float rr = sigm(gir + ghr);
    float zz = sigm(giz + ghz);
    float nn = tanhf(gin + rr * ghn);
    float hold = hin[(size_t)rowi * 128 + j];
    float hnew = (1.0f - zz) * nn + zz * hold;
    s_h[mr * 128 + j] = hnew;
    hout[(size_t)rowi * 128 + j] = hnew;
  }
  __syncthreads();

  // ---- heads: 16 rows x 6 outputs, dot over 128 ----
  if (tid < 96) {
    int mr = tid / 6, j = tid % 6;
    const float* wrow;
    float bias;
    switch (j) {
      case 0: wrow = pw;        bias = pb[0]; break;
      case 1: wrow = mw;        bias = mb[0]; break;
      case 2: wrow = mw + 128;  bias = mb[1]; break;
      case 3: wrow = lw;        bias = lb[0]; break;
      case 4: wrow = lw + 128;  bias = lb[1]; break;
      default: wrow = cw;       bias = cb[0]; break;
    }
    float d = bias;
#pragma unroll 8
    for (int k = 0; k < 128; ++k) d += s_h[mr * 128 + k] * wrow[k];
    s_head[mr * 6 + j] = d;
  }
  __syncthreads();

  // ---- finalize row: outputs + sampled a_{s+1} ----
  if (tid < 16) {
    int rowi = r0 + tid;
    float d0 = s_head[tid * 6 + 0];  // log_pi
    float d1 = s_head[tid * 6 + 1];  // mu_x
    float d2 = s_head[tid * 6 + 2];  // mu_y
    float d3 = s_head[tid * 6 + 3];  // log_sigma_x
    float d4 = s_head[tid * 6 + 4];  // log_sigma_y
    float rho = tanhf(s_head[tid * 6 + 5]);
    float* o = out + ((size_t)step * NB + rowi) * 6;
    o[0] = d0; o[1] = d1; o[2] = d2; o[3] = d3; o[4] = d4; o[5] = rho;
    float e0 = eps[((size_t)step * NB + rowi) * 2 + 0];
    float e1 = eps[((size_t)step * NB + rowi) * 2 + 1];
    float sx = expf(d3), sy = expf(d4);
    float ax = d1 + sx * e0;
    float ay = d2 + sy * (rho * e0 + sqrtf(fmaxf(1.0f - rho * rho, 1e-12f)) * e1);
    inp[(size_t)rowi * INP_STRIDE + 180] = ax;
    inp[(size_t)rowi * INP_STRIDE + 181] = ay;
  }
}

// ---------------------------------------------------------------------------
extern "C" void kernel_launch(void* const* d_in, const int* in_sizes, int n_in,
                              void* d_out, int out_size, void* d_ws,
                              size_t ws_size, hipStream_t stream) {
  const float* scene = (const float*)d_in[0];
  const float* eps   = (const float*)d_in[1];
  const float* nw_f  = (const float*)d_in[2];
  const float* nu_f  = (const float*)d_in[3];
  const float* nbi_f = (const float*)d_in[4];
  const float* nbh_f = (const float*)d_in[5];
  const float* nw_b  = (const float*)d_in[6];
  const float* nu_b  = (const float*)d_in[7];
  const float* nbi_b = (const float*)d_in[8];
  const float* nbh_b = (const float*)d_in[9];
  const float* ew_f  = (const float*)d_in[10];
  const float* eu_f  = (const float*)d_in[11];
  const float* ebi_f = (const float*)d_in[12];
  const float* ebh_f = (const float*)d_in[13];
  const float* ew_b  = (const float*)d_in[14];
  const float* eu_b  = (const float*)d_in[15];
  const float* ebi_b = (const float*)d_in[16];
  const float* ebh_b = (const float*)d_in[17];
  const float* gw    = (const float*)d_in[18];
  const float* gu    = (const float*)d_in[19];
  const float* gbi   = (const float*)d_in[20];
  const float* gbh   = (const float*)d_in[21];
  const float* aw    = (const float*)d_in[22];
  const float* ab    = (const float*)d_in[23];
  const float* sw    = (const float*)d_in[24];
  const float* sb    = (const float*)d_in[25];
  const float* pw    = (const float*)d_in[26];
  const float* pb    = (const float*)d_in[27];
  const float* mw    = (const float*)d_in[28];
  const float* mb    = (const float*)d_in[29];
  const float* lw    = (const float*)d_in[30];
  const float* lb    = (const float*)d_in[31];
  const float* cw    = (const float*)d_in[32];
  const float* cb    = (const float*)d_in[33];
  float* out = (float*)d_out;

  float* ws = (float*)d_ws;
  float* inp    = ws;                       // 1024 * 192
  float* stateA = inp + NB * INP_STRIDE;    // 1024 * 128
  float* stateB = stateA + NB * 128;        // 1024 * 128
  float* gwp    = stateB + NB * 128;        // 384 * 184

  pad_gw_kernel<<<(384 * GWP_STRIDE + 255) / 256, 256, 0, stream>>>(gw, gwp);
  scene_kernel<<<4, 256, 0, stream>>>(scene, nw_f, nu_f, nbi_f, nbh_f, nw_b,
                                      nu_b, nbi_b, nbh_b, aw, ab, inp);
  edge_kernel<<<4, 256, 0, stream>>>(scene, ew_f, eu_f, ebi_f, ebh_f, ew_b,
                                     eu_b, ebi_b, ebh_b, inp);
  state0_kernel<<<64, 256, 0, stream>>>(inp, sw, sb, stateA);

  for (int s = 0; s < 12; ++s) {
    float* hin = (s & 1) ? stateB : stateA;
    float* hout = (s & 1) ? stateA : stateB;
    gru_step_kernel<<<64, 384, 0, stream>>>(inp, gwp, gu, gbi, gbh, hin, hout,
                                            pw, pb, mw, mb, lw, lb, cw, cb,
                                            eps, out, s);
  }
}